// AttnDecoderRNN2_4913442586940
// MI455X (gfx1250) — compile-verified
//
#include <hip/hip_runtime.h>
#include <hip/hip_bf16.h>
#include <math.h>

#define B_    128
#define H_    1024
#define EMB_  512
#define ATTN_ 1024
#define V_    50257
#define IL_   512
#define H2_   2048
#define H4_   4096
#define CH_   8            // attention L-chunks
#define LC_   (IL_ / CH_)  // 64 rows per chunk

typedef __attribute__((ext_vector_type(2))) float v2f;
typedef __attribute__((ext_vector_type(8))) float v8f;

__device__ __forceinline__ float sigmoidf_(float x) { return 1.0f / (1.0f + expf(-x)); }

#define WMMA_F32(a, b, c) \
  __builtin_amdgcn_wmma_f32_16x16x4_f32(false, (a), false, (b), (short)0, (c), false, false)

// ---------------------------------------------------------------------------
// Embedding gather with padding_idx=0
// ---------------------------------------------------------------------------
__global__ void embed_kernel(const int* __restrict__ ids,
                             const float* __restrict__ emb,
                             float* __restrict__ out) {
  int idx = blockIdx.x * blockDim.x + threadIdx.x;           // B*EMB
  if (idx >= B_ * EMB_) return;
  int b = idx / EMB_;
  int e = idx - b * EMB_;
  int id = ids[b];
  out[idx] = (id != 0) ? emb[(size_t)id * EMB_ + e] : 0.0f;
}

// ---------------------------------------------------------------------------
// WMMA core: 4 M-tiles (64 rows) x one 16-col weight strip, software-
// pipelined in 8-K chunks: loads for chunk k+8 are issued BEFORE the WMMAs
// consuming chunk k, so global latency overlaps the matrix pipe in-wave.
// xbase = X + (m0+lo)*ldx + 2*hi ; wp = W + wrow*ldw + 2*hi ; K % 8 == 0.
// ---------------------------------------------------------------------------
template <bool MASKED>
__device__ __forceinline__ void wmma_acc_4m(v8f acc[4],
                                            const float* __restrict__ xbase,
                                            int ldx16,
                                            const float* __restrict__ wp,
                                            int K, float nmsk) {
  v2f cb[2], ca0[2], ca1[2], ca2[2], ca3[2];
#pragma unroll
  for (int u = 0; u < 2; ++u) {                    // preload chunk at k=0
    cb[u]  = *(const v2f*)(wp + 4 * u);
    ca0[u] = *(const v2f*)(xbase + 4 * u);
    ca1[u] = *(const v2f*)(xbase + ldx16 + 4 * u);
    ca2[u] = *(const v2f*)(xbase + 2 * ldx16 + 4 * u);
    ca3[u] = *(const v2f*)(xbase + 3 * ldx16 + 4 * u);
  }
  int k = 0;
#pragma unroll 2
  for (; k < K - 8; k += 8) {
    v2f nb[2], na0[2], na1[2], na2[2], na3[2];
#pragma unroll
    for (int u = 0; u < 2; ++u) {                  // prefetch chunk k+8
      const float* wq = wp + k + 8 + 4 * u;
      const float* xq = xbase + k + 8 + 4 * u;
      nb[u]  = *(const v2f*)(wq);
      na0[u] = *(const v2f*)(xq);
      na1[u] = *(const v2f*)(xq + ldx16);
      na2[u] = *(const v2f*)(xq + 2 * ldx16);
      na3[u] = *(const v2f*)(xq + 3 * ldx16);
    }
#pragma unroll
    for (int u = 0; u < 2; ++u) {                  // consume chunk k
      v2f bb = cb[u];
      if (MASKED) bb *= nmsk;
      acc[0] = WMMA_F32(ca0[u], bb, acc[0]);
      acc[1] = WMMA_F32(ca1[u], bb, acc[1]);
      acc[2] = WMMA_F32(ca2[u], bb, acc[2]);
      acc[3] = WMMA_F32(ca3[u], bb, acc[3]);
    }
#pragma unroll
    for (int u = 0; u < 2; ++u) {                  // rotate buffers
      cb[u] = nb[u];
      ca0[u] = na0[u]; ca1[u] = na1[u]; ca2[u] = na2[u]; ca3[u] = na3[u];
    }
  }
#pragma unroll
  for (int u = 0; u < 2; ++u) {                    // peeled final chunk
    v2f bb = cb[u];
    if (MASKED) bb *= nmsk;
    acc[0] = WMMA_F32(ca0[u], bb, acc[0]);
    acc[1] = WMMA_F32(ca1[u], bb, acc[1]);
    acc[2] = WMMA_F32(ca2[u], bb, acc[2]);
    acc[3] = WMMA_F32(ca3[u], bb, acc[3]);
  }
}

// ---------------------------------------------------------------------------
// D = X1 @ W1^T [+ X2 @ W2^T] [+ b1] [+ b2], optional tanh epilogue.
// X: (M,K) row-major, W: (N,K) row-major. One wave -> 64x16 of D (4 M-tiles),
// 4 waves/block -> 64x64 per block. M must be a multiple of 64 (M=128 here).
// ---------------------------------------------------------------------------
__global__ void gemm2_bias_wmma(const float* __restrict__ X1, const float* __restrict__ W1, int K1,
                                const float* __restrict__ X2, const float* __restrict__ W2, int K2,
                                const float* __restrict__ b1, const float* __restrict__ b2,
                                float* __restrict__ D, int M, int N, int act) {
  const int lane = threadIdx.x & 31;
  const int wave = threadIdx.x >> 5;
  const int lo   = lane & 15;
  const int hi   = lane >> 4;
  const int m0   = blockIdx.y * 64;
  const int n0   = (blockIdx.x * (blockDim.x >> 5) + wave) * 16;
  if (n0 >= N || m0 >= M) return;                            // wave-uniform

  const int  ncand = n0 + lo;                                // W row = output col
  const bool full  = (n0 + 16 <= N);                         // wave-uniform
  const int  wrow  = (ncand < N) ? ncand : (N - 1);
  const float nmsk = (ncand < N) ? 1.0f : 0.0f;

  v8f z = {0.f, 0.f, 0.f, 0.f, 0.f, 0.f, 0.f, 0.f};
  v8f acc[4] = {z, z, z, z};

  if (full) {
    wmma_acc_4m<false>(acc, X1 + (size_t)(m0 + lo) * K1 + 2 * hi, 16 * K1,
                       W1 + (size_t)wrow * K1 + 2 * hi, K1, 1.0f);
    if (X2)
      wmma_acc_4m<false>(acc, X2 + (size_t)(m0 + lo) * K2 + 2 * hi, 16 * K2,
                         W2 + (size_t)wrow * K2 + 2 * hi, K2, 1.0f);
  } else {
    wmma_acc_4m<true>(acc, X1 + (size_t)(m0 + lo) * K1 + 2 * hi, 16 * K1,
                      W1 + (size_t)wrow * K1 + 2 * hi, K1, nmsk);
    if (X2)
      wmma_acc_4m<true>(acc, X2 + (size_t)(m0 + lo) * K2 + 2 * hi, 16 * K2,
                        W2 + (size_t)wrow * K2 + 2 * hi, K2, nmsk);
  }

  float bias = 0.0f;
  if (b1) bias += b1[wrow];
  if (b2) bias += b2[wrow];

  if (ncand < N) {
#pragma unroll
    for (int tm = 0; tm < 4; ++tm) {
#pragma unroll
      for (int r = 0; r < 8; ++r) {
        int row = m0 + 16 * tm + r + 8 * hi;
        float v = acc[tm][r] + bias;
        if (act == 1) v = tanhf(v);
        D[(size_t)row * N + ncand] = v;
      }
    }
  }
}

// ---------------------------------------------------------------------------
// LSTM pointwise: gates (B,4H) in [i,f,g,o] chunk order -> h_new, c_new.
// Optionally mirrors h/c into a strided buffer (builds dec = [h2,c2]).
// ---------------------------------------------------------------------------
__global__ void lstm_pointwise(const float* __restrict__ gates,
                               const float* __restrict__ c_in,
                               float* __restrict__ h_out, float* __restrict__ c_out,
                               float* __restrict__ h_copy, float* __restrict__ c_copy,
                               int copy_stride) {
  int idx = blockIdx.x * blockDim.x + threadIdx.x;           // B*H
  if (idx >= B_ * H_) return;
  int b = idx >> 10;                                         // H_ = 1024
  int h = idx & (H_ - 1);
  const float* g = gates + (size_t)b * H4_;
  float gi = sigmoidf_(g[h]);
  float gf = sigmoidf_(g[H_ + h]);
  float gg = tanhf(g[2 * H_ + h]);
  float go = sigmoidf_(g[3 * H_ + h]);
  float c  = gf * c_in[idx] + gi * gg;
  float hn = go * tanhf(c);
  h_out[idx] = hn;
  c_out[idx] = c;
  if (h_copy) h_copy[(size_t)b * copy_stride + h] = hn;
  if (c_copy) c_copy[(size_t)b * copy_stride + h] = c;
}

// ---------------------------------------------------------------------------
// Fused attention pass 1 (flash-style, split over L):
// one block per (chunk,b) streams enc ONCE, computing raw scores and
// online-softmax partials (m, den, ctx[2048]) for its 64 L-rows.
// Each thread owns d in [t*8, t*8+8) -> float4-pair loads/stores.
// ---------------------------------------------------------------------------
__global__ void attn_fused_pass1(const float* __restrict__ enc,
                                 const float* __restrict__ score,
                                 float* __restrict__ scores_raw,
                                 float* __restrict__ ctx_part,
                                 float* __restrict__ m_part,
                                 float* __restrict__ d_part) {
  __shared__ float red[8];
  const int b = blockIdx.y, chunk = blockIdx.x, t = threadIdx.x;  // 256 threads
  const int lane = t & 31, wid = t >> 5;
  const int d0 = t * 8;

  const float* srow = score + (size_t)b * H2_;
  float4 s0 = *(const float4*)(srow + d0);
  float4 s1 = *(const float4*)(srow + d0 + 4);

  float4 cx0 = {0.f, 0.f, 0.f, 0.f}, cx1 = {0.f, 0.f, 0.f, 0.f};
  float mrun = -INFINITY, den = 0.0f;

  const int l0 = chunk * LC_;
  for (int li = 0; li < LC_; ++li) {
    const int l = l0 + li;
    const float* erow = enc + ((size_t)l * B_ + b) * H2_;
    float4 e0 = *(const float4*)(erow + d0);
    float4 e1 = *(const float4*)(erow + d0 + 4);
    float partial = e0.x * s0.x + e0.y * s0.y + e0.z * s0.z + e0.w * s0.w +
                    e1.x * s1.x + e1.y * s1.y + e1.z * s1.z + e1.w * s1.w;
    for (int off = 16; off; off >>= 1) partial += __shfl_xor(partial, off, 32);
    if (lane == 0) red[wid] = partial;
    __syncthreads();
    float sc = red[0] + red[1] + red[2] + red[3] + red[4] + red[5] + red[6] + red[7];
    __syncthreads();
    if (t == 0) scores_raw[(size_t)b * IL_ + l] = sc;

    float mn    = fmaxf(mrun, sc);
    float scale = expf(mrun - mn);                 // 0 on first iteration
    float w     = expf(sc - mn);
    den = den * scale + w;
    cx0.x = cx0.x * scale + w * e0.x;  cx0.y = cx0.y * scale + w * e0.y;
    cx0.z = cx0.z * scale + w * e0.z;  cx0.w = cx0.w * scale + w * e0.w;
    cx1.x = cx1.x * scale + w * e1.x;  cx1.y = cx1.y * scale + w * e1.y;
    cx1.z = cx1.z * scale + w * e1.z;  cx1.w = cx1.w * scale + w * e1.w;
    mrun = mn;
  }

  float* cp = ctx_part + ((size_t)chunk * B_ + b) * H2_;
  *(float4*)(cp + d0)     = cx0;
  *(float4*)(cp + d0 + 4) = cx1;
  if (t == 0) {
    m_part[b * CH_ + chunk] = mrun;
    d_part[b * CH_ + chunk] = den;
  }
}

// Combine chunk partials -> global (m, 1/den) + normalized attn weights.
__global__ void attn_combine_weights(const float* __restrict__ m_part,
                                     const float* __restrict__ d_part,
                                     const float* __restrict__ scores_raw,
                                     float* __restrict__ attn_out,
                                     float* __restrict__ gm, float* __restrict__ gdinv) {
  int b = blockIdx.x, t = threadIdx.x;                       // 256 threads
  float m = -INFINITY;
#pragma unroll
  for (int c = 0; c < CH_; ++c) m = fmaxf(m, m_part[b * CH_ + c]);
  float den = 0.0f;
#pragma unroll
  for (int c = 0; c < CH_; ++c) den += d_part[b * CH_ + c] * expf(m_part[b * CH_ + c] - m);
  float inv = 1.0f / den;
  if (t == 0) { gm[b] = m; gdinv[b] = inv; }
  for (int l = t; l < IL_; l += 256)
    attn_out[(size_t)b * IL_ + l] = expf(scores_raw[(size_t)b * IL_ + l] - m) * inv;
}

// Combine chunk context partials -> concat[:, 0:2H]
__global__ void attn_combine_context(const float* __restrict__ ctx_part,
                                     const float* __restrict__ m_part,
                                     const float* __restrict__ gm,
                                     const float* __restrict__ gdinv,
                                     float* __restrict__ concatb) {
  int idx = blockIdx.x * blockDim.x + threadIdx.x;           // B*2H
  if (idx >= B_ * H2_) return;
  int b = idx >> 11;                                         // H2_ = 2048
  int d = idx & (H2_ - 1);
  float m = gm[b], inv = gdinv[b];
  float acc = 0.0f;
#pragma unroll
  for (int c = 0; c < CH_; ++c)
    acc += ctx_part[((size_t)c * B_ + b) * H2_ + d] * expf(m_part[b * CH_ + c] - m);
  concatb[(size_t)b * H4_ + d] = acc * inv;
}

__global__ void copy_dec_to_concat(const float* __restrict__ dec,
                                   float* __restrict__ concatb) {
  int idx = blockIdx.x * blockDim.x + threadIdx.x;           // B*2H
  if (idx >= B_ * H2_) return;
  int b = idx >> 11;
  int j = idx & (H2_ - 1);
  concatb[(size_t)b * H4_ + H2_ + j] = dec[idx];
}

// ---------------------------------------------------------------------------
// In-place log_softmax over V per batch row.
// ---------------------------------------------------------------------------
__global__ void log_softmax_v(float* __restrict__ logits) {
  __shared__ float red[1024];
  int b = blockIdx.x, t = threadIdx.x;                       // 1024 threads
  float* row = logits + (size_t)b * V_;
  float m = -INFINITY;
  for (int i = t; i < V_; i += 1024) m = fmaxf(m, row[i]);
  red[t] = m; __syncthreads();
  for (int s = 512; s; s >>= 1) { if (t < s) red[t] = fmaxf(red[t], red[t + s]); __syncthreads(); }
  m = red[0]; __syncthreads();
  float sum = 0.0f;
  for (int i = t; i < V_; i += 1024) sum += expf(row[i] - m);
  red[t] = sum; __syncthreads();
  for (int s = 512; s; s >>= 1) { if (t < s) red[t] += red[t + s]; __syncthreads(); }
  float lse = m + logf(red[0]);
  for (int i = t; i < V_; i += 1024) row[i] = row[i] - lse;
}

// ---------------------------------------------------------------------------
extern "C" void kernel_launch(void* const* d_in, const int* in_sizes, int n_in,
                              void* d_out, int out_size, void* d_ws, size_t ws_size,
                              hipStream_t stream) {
  const int*   ids     = (const int*)d_in[0];
  const float* h1      = (const float*)d_in[1];
  const float* c1      = (const float*)d_in[2];
  const float* h2      = (const float*)d_in[3];
  const float* c2      = (const float*)d_in[4];
  const float* enc     = (const float*)d_in[5];
  const float* embT    = (const float*)d_in[6];
  const float* Wih1    = (const float*)d_in[7];
  const float* Whh1    = (const float*)d_in[8];
  const float* bih1    = (const float*)d_in[9];
  const float* bhh1    = (const float*)d_in[10];
  const float* Wih2    = (const float*)d_in[11];
  const float* Whh2    = (const float*)d_in[12];
  const float* bih2    = (const float*)d_in[13];
  const float* bhh2    = (const float*)d_in[14];
  const float* W_score = (const float*)d_in[15];
  const float* b_score = (const float*)d_in[16];
  const float* W_attn  = (const float*)d_in[17];
  const float* b_attn  = (const float*)d_in[18];
  const float* W_out   = (const float*)d_in[19];
  const float* b_out   = (const float*)d_in[20];

  float* out        = (float*)d_out;
  float* out_logits = out;                                   // B*V
  float* out_h1     = out_logits + (size_t)B_ * V_;          // B*H
  float* out_c1     = out_h1 + (size_t)B_ * H_;
  float* out_h2     = out_c1 + (size_t)B_ * H_;
  float* out_c2     = out_h2 + (size_t)B_ * H_;
  float* out_attn   = out_c2 + (size_t)B_ * H_;              // B*IL

  float* ws          = (float*)d_ws;
  float* embedded    = ws;                                   // B*EMB
  float* gates       = embedded + (size_t)B_ * EMB_;         // B*4H (reused)
  float* dec         = gates + (size_t)B_ * H4_;             // B*2H
  float* score       = dec + (size_t)B_ * H2_;               // B*2H
  float* scores_raw  = score + (size_t)B_ * H2_;             // B*IL
  float* concatb     = scores_raw + (size_t)B_ * IL_;        // B*4H
  float* attentional = concatb + (size_t)B_ * H4_;           // B*ATTN
  float* ctx_part    = attentional + (size_t)B_ * ATTN_;     // CH*B*2H
  float* m_part      = ctx_part + (size_t)CH_ * B_ * H2_;    // CH*B
  float* d_part      = m_part + CH_ * B_;                    // CH*B
  float* gm          = d_part + CH_ * B_;                    // B
  float* gdinv       = gm + B_;                              // B

  // 1) embedding
  embed_kernel<<<(B_ * EMB_ + 255) / 256, 256, 0, stream>>>(ids, embT, embedded);

  // 2) LSTM cell 1: gates = emb@Wih1^T + h1@Whh1^T + bih1 + bhh1
  gemm2_bias_wmma<<<dim3(H4_ / 64, B_ / 64), 128, 0, stream>>>(
      embedded, Wih1, EMB_, h1, Whh1, H_, bih1, bhh1, gates, B_, H4_, 0);
  lstm_pointwise<<<(B_ * H_ + 255) / 256, 256, 0, stream>>>(
      gates, c1, out_h1, out_c1, nullptr, nullptr, 0);

  // 3) LSTM cell 2 (also builds dec = [h2n, c2n] in ws)
  gemm2_bias_wmma<<<dim3(H4_ / 64, B_ / 64), 128, 0, stream>>>(
      out_h1, Wih2, H_, h2, Whh2, H_, bih2, bhh2, gates, B_, H4_, 0);
  lstm_pointwise<<<(B_ * H_ + 255) / 256, 256, 0, stream>>>(
      gates, c2, out_h2, out_c2, dec, dec + H_, H2_);

  // 4) score = dec @ W_score^T + b_score
  gemm2_bias_wmma<<<dim3(H2_ / 64, B_ / 64), 128, 0, stream>>>(
      dec, W_score, H2_, nullptr, nullptr, 0, b_score, nullptr, score, B_, H2_, 0);

  // 5) fused attention: single pass over enc (scores + online softmax + ctx)
  attn_fused_pass1<<<dim3(CH_, B_), 256, 0, stream>>>(
      enc, score, scores_raw, ctx_part, m_part, d_part);
  attn_combine_weights<<<B_, 256, 0, stream>>>(
      m_part, d_part, scores_raw, out_attn, gm, gdinv);
  attn_combine_context<<<(B_ * H2_ + 255) / 256, 256, 0, stream>>>(
      ctx_part, m_part, gm, gdinv, concatb);
  copy_dec_to_concat<<<(B_ * H2_ + 255) / 256, 256, 0, stream>>>(dec, concatb);

  // 6) attentional = tanh(concat @ W_attn^T + b_attn)
  gemm2_bias_wmma<<<dim3(ATTN_ / 64, B_ / 64), 128, 0, stream>>>(
      concatb, W_attn, H4_, nullptr, nullptr, 0, b_attn, nullptr,
      attentional, B_, ATTN_, 1);

  // 7) logits = attentional @ W_out^T + b_out   (N=50257, tail-guarded)
  gemm2_bias_wmma<<<dim3((V_ + 63) / 64, B_ / 64), 128, 0, stream>>>(
      attentional, W_out, ATTN_, nullptr, nullptr, 0, b_out, nullptr,
      out_logits, B_, V_, 0);

  // 8) log_softmax in place on d_out
  log_softmax_v<<<B_, 1024, 0, stream>>>(out_logits);
}